// Propagation_8349416424063
// MI455X (gfx1250) — compile-verified
//
#include <hip/hip_runtime.h>
#include <stdint.h>

#define BB 4
#define DD 8
#define HH 512
#define WW 640
#define NN 16
#define NPROP 24      // DD + NN live elements
#define NNET 32       // Batcher network generated for 32, pad comparators dropped
#define TPB 256       // 8 wave32 waves per block

#define HW (HH * WW)                 // 327,680 pixels per plane
#define GPLANE_BYTES (HW * 8)        // 2,621,440 B per (b,n) grid plane (float2)

// ---------------------------------------------------------------------------
// CDNA5 async global->LDS copy, GVS (saddr) addressing mode:
//   mem_addr = SGPR64 base + VGPR32 byte offset + signed imm24
// VDST VGPR holds the wave-relative LDS byte address (low 32 bits of a
// generic pointer to a __shared__ object). Tracked by ASYNCcnt.
// ---------------------------------------------------------------------------
template <int IMM>
__device__ __forceinline__ void async_copy_b64_saddr(void* lds_ptr,
                                                     const float* sbase,
                                                     uint32_t voff_bytes) {
    uint32_t lds_addr = (uint32_t)(uintptr_t)lds_ptr;
    asm volatile("global_load_async_to_lds_b64 %0, %1, %2 offset:%3"
                 :
                 : "v"(lds_addr), "v"(voff_bytes),
                   "s"((unsigned long long)(uintptr_t)sbase), "i"(IMM)
                 : "memory");
}

__device__ __forceinline__ void wait_asynccnt0() {
    asm volatile("s_wait_asynccnt 0" ::: "memory");
}

__global__ __launch_bounds__(TPB) void prop_sort_kernel(
    const float* __restrict__ depth,   // [B, D, H, W]
    const float* __restrict__ grid,    // [B, N*H, W, 2]
    float* __restrict__ out)           // [B, 24, H, W] sorted along dim 1
{
    __shared__ float2 gbuf[NN * TPB];  // 32 KB staged grid coordinates

    const int tx = threadIdx.x;
    const int b  = blockIdx.y;                     // batch (uniform per block)
    const int yx = blockIdx.x * TPB + tx;          // pixel within image (exact)
    const int y  = yx / WW;
    const int x  = yx - y * WW;

    // Uniform per-batch base pointers (SGPRs); all per-lane offsets are i32.
    const float* dbase  = depth + (size_t)b * (DD * HW);
    const float* center = dbase + (DD / 2) * HW;
    const float* gbase  = grid + (size_t)b * ((size_t)NN * HW * 2);
    float*       obase  = out + (size_t)b * (NPROP * HW);

    // ---- stage 16 grid float2's per pixel into LDS asynchronously ----------
    // Neighbor n source byte offset = yx*8 + n*GPLANE_BYTES. Fold n%4 strides
    // into the signed imm24 field (max 7,864,320 < 2^23), one v_add per group.
    const uint32_t voff0 = (uint32_t)yx * 8u;
#pragma unroll
    for (int g = 0; g < 4; ++g) {
        uint32_t voff = voff0 + (uint32_t)g * (4u * GPLANE_BYTES);
        async_copy_b64_saddr<0 * GPLANE_BYTES>(&gbuf[(g * 4 + 0) * TPB + tx], gbase, voff);
        async_copy_b64_saddr<1 * GPLANE_BYTES>(&gbuf[(g * 4 + 1) * TPB + tx], gbase, voff);
        async_copy_b64_saddr<2 * GPLANE_BYTES>(&gbuf[(g * 4 + 2) * TPB + tx], gbase, voff);
        async_copy_b64_saddr<3 * GPLANE_BYTES>(&gbuf[(g * 4 + 3) * TPB + tx], gbase, voff);
    }

    float v[NPROP];

    // ---- overlap: pull the 8 depth hypotheses while async loads fly --------
#pragma unroll
    for (int d = 0; d < DD; ++d)
        v[d] = dbase[yx + d * HW];                 // saddr + i32 voffset

    wait_asynccnt0();   // drain ASYNCcnt -> LDS slots valid for this wave

    // ---- 16 bilinear samples (border padding, align_corners=False) ---------
#pragma unroll
    for (int n = 0; n < NN; ++n) {
        float2 g = gbuf[n * TPB + tx];
        float fx = (g.x + 1.0f) * (0.5f * (float)WW) - 0.5f;
        float fy = (g.y + 1.0f) * (0.5f * (float)HH) - 0.5f;
        fx = fminf(fmaxf(fx, 0.0f), (float)(WW - 1));
        fy = fminf(fmaxf(fy, 0.0f), (float)(HH - 1));
        float x0f = floorf(fx), y0f = floorf(fy);
        float wx = fx - x0f, wy = fy - y0f;
        int x0 = (int)x0f, y0 = (int)y0f;
        int x1 = (x0 + 1 < WW) ? x0 + 1 : WW - 1;
        int y1 = (y0 + 1 < HH) ? y0 + 1 : HH - 1;
        int r0 = y0 * WW, r1 = y1 * WW;            // i32 gather offsets
        float v00 = center[r0 + x0], v01 = center[r0 + x1];
        float v10 = center[r1 + x0], v11 = center[r1 + x1];
        float top = v00 * (1.0f - wx) + v01 * wx;
        float bot = v10 * (1.0f - wx) + v11 * wx;
        v[DD + n] = top * (1.0f - wy) + bot * wy;
    }

    // ---- Batcher odd-even mergesort network for 32, restricted to the 24
    //      live indices. All comparators are ascending, so +inf sentinels at
    //      indices 24..31 make every comparator touching index >= 24 an
    //      identity; the b2 < NPROP guard drops them at compile time.
    //      ~135 compare-exchanges, fully unrolled, all indices constant.
#pragma unroll
    for (int p = 1; p < NNET; p <<= 1) {
#pragma unroll
        for (int k = p; k >= 1; k >>= 1) {
#pragma unroll
            for (int j = k % p; j + k < NNET; j += 2 * k) {
#pragma unroll
                for (int i = 0; i < k; ++i) {
                    int a  = i + j;
                    int b2 = i + j + k;
                    if (b2 < NPROP && (a / (2 * p)) == (b2 / (2 * p))) {
                        float lo = fminf(v[a], v[b2]);
                        float hi = fmaxf(v[a], v[b2]);
                        v[a]  = lo;
                        v[b2] = hi;
                    }
                }
            }
        }
    }

    // ---- write 24 sorted values, coalesced per plane -----------------------
#pragma unroll
    for (int i = 0; i < NPROP; ++i)
        obase[yx + i * HW] = v[i];                 // saddr + i32 voffset
}

extern "C" void kernel_launch(void* const* d_in, const int* in_sizes, int n_in,
                              void* d_out, int out_size, void* d_ws, size_t ws_size,
                              hipStream_t stream) {
    // setup_inputs order: batch(0), height(1), width(2), depth_sample(3),
    // grid(4), depth_min(5), depth_max(6), depth_interval_scale(7).
    // depth_min/max/scale are unused by the reference computation.
    const float* depth = (const float*)d_in[3];
    const float* grid  = (const float*)d_in[4];
    float* out = (float*)d_out;

    dim3 gridDim(HW / TPB, BB, 1);                  // (1280, 4) blocks, exact
    prop_sort_kernel<<<gridDim, TPB, 0, stream>>>(depth, grid, out);
}